// AttentionFusion_30073361007255
// MI455X (gfx1250) — compile-verified
//
#include <hip/hip_runtime.h>

typedef __attribute__((ext_vector_type(2))) float v2f;
typedef __attribute__((ext_vector_type(8))) float v8f;

#define DIM 64          // feature dim
#define KCH 16          // 64 / K=4 chunks

// One wave32 handles a 16-row tile:
//   lane l: row r = l%16, K-subpair h = l>>4 (K offsets 2h,2h+1 of each chunk)
//   B = X^T chunk (b64 per lane), A = w broadcast (preloaded), C row = scores.
__global__ __launch_bounds__(256) void
attn_fuse_wmma(const float* __restrict__ x1, const float* __restrict__ x2,
               const float* __restrict__ w, float* __restrict__ out,
               int n, int ntiles)
{
    const int lane = threadIdx.x & 31;
    const int r    = lane & 15;   // row within tile
    const int h    = lane >> 4;   // K-subpair selector
    const int wavesPerBlock = blockDim.x >> 5;
    const int waveId = blockIdx.x * wavesPerBlock + (threadIdx.x >> 5);
    const int nWaves = gridDim.x * wavesPerBlock;

    // Preload w in A-matrix layout: chunk c -> lane needs w[4c+2h], w[4c+2h+1]
    v2f aw[KCH];
    #pragma unroll
    for (int c = 0; c < KCH; ++c)
        aw[c] = *(const v2f*)(w + 4 * c + 2 * h);

    for (int tile = waveId; tile < ntiles; tile += nWaves) {
        const int rowBase = tile * 16;

        // Prefetch next tile this wave will touch (covers full 256B rows via h).
        const int nextTile = tile + nWaves;
        if (nextTile < ntiles) {
            const size_t nb = (size_t)(nextTile * 16 + r) * DIM + (size_t)h * 32;
            __builtin_prefetch(x1 + nb, 0, 1);
            __builtin_prefetch(x2 + nb, 0, 1);
        }

        if (rowBase + 16 <= n) {
            const float* p1 = x1 + (size_t)(rowBase + r) * DIM + 2 * h;
            const float* p2 = x2 + (size_t)(rowBase + r) * DIM + 2 * h;

            // Issue all loads up front for deep memory-level parallelism.
            v2f b1[KCH], b2[KCH];
            #pragma unroll
            for (int c = 0; c < KCH; ++c) b1[c] = *(const v2f*)(p1 + 4 * c);
            #pragma unroll
            for (int c = 0; c < KCH; ++c) b2[c] = *(const v2f*)(p2 + 4 * c);

            // Dot products on the matrix pipe: acc[i][j] = <w, x[row j]>
            v8f acc1 = {}; v8f acc2 = {};
            #pragma unroll
            for (int c = 0; c < KCH; ++c) {
                acc1 = __builtin_amdgcn_wmma_f32_16x16x4_f32(
                           false, aw[c], false, b1[c], (short)0, acc1, false, false);
                acc2 = __builtin_amdgcn_wmma_f32_16x16x4_f32(
                           false, aw[c], false, b2[c], (short)0, acc2, false, false);
            }
            // C VGPR0: lanes0-15 -> (M=0,N=l), lanes16-31 -> (M=8,N=l-16);
            // all C rows identical, so acc[0] = s[r] in every lane.
            const float s1 = acc1[0];
            const float s2 = acc2[0];
            const float a1 = 1.0f / (1.0f + __expf(s2 - s1));  // sigmoid(s1-s2)
            const float a2 = 1.0f - a1;

            // Blend is lane-local on the same registers; store b64 per chunk.
            float* po = out + (size_t)(rowBase + r) * DIM + 2 * h;
            #pragma unroll
            for (int c = 0; c < KCH; ++c) {
                v2f f;
                f.x = a1 * b1[c].x + a2 * b2[c].x;
                f.y = a1 * b1[c].y + a2 * b2[c].y;
                *(v2f*)(po + 4 * c) = f;
            }
            // alpha [N,2] appended after fused [N,64]; one lane-half writes.
            if (lane < 16) {
                v2f al; al.x = a1; al.y = a2;
                *(v2f*)(out + (size_t)n * DIM + (size_t)(rowBase + r) * 2) = al;
            }
        } else {
            // Tail (never hit for N=1M): scalar per-row, no WMMA under divergence.
            const int row = rowBase + lane;
            if (row < n) {
                float s1 = 0.f, s2 = 0.f;
                for (int k = 0; k < DIM; ++k) {
                    s1 += x1[(size_t)row * DIM + k] * w[k];
                    s2 += x2[(size_t)row * DIM + k] * w[k];
                }
                const float a1 = 1.0f / (1.0f + __expf(s2 - s1));
                const float a2 = 1.0f - a1;
                for (int k = 0; k < DIM; ++k)
                    out[(size_t)row * DIM + k] =
                        a1 * x1[(size_t)row * DIM + k] + a2 * x2[(size_t)row * DIM + k];
                out[(size_t)n * DIM + (size_t)row * 2 + 0] = a1;
                out[(size_t)n * DIM + (size_t)row * 2 + 1] = a2;
            }
        }
    }
}

extern "C" void kernel_launch(void* const* d_in, const int* in_sizes, int n_in,
                              void* d_out, int out_size, void* d_ws, size_t ws_size,
                              hipStream_t stream) {
    const float* x1 = (const float*)d_in[0];
    const float* x2 = (const float*)d_in[1];
    const float* w  = (const float*)d_in[2];
    float* out = (float*)d_out;

    const int n = in_sizes[0] / DIM;          // rows
    const int ntiles = (n + 15) / 16;
    const int threads = 256;                  // 8 waves/WG on wave32
    const int wavesPerBlock = threads / 32;
    int blocks = (ntiles + wavesPerBlock - 1) / wavesPerBlock;
    if (blocks < 1) blocks = 1;

    attn_fuse_wmma<<<blocks, threads, 0, stream>>>(x1, x2, w, out, n, ntiles);
}